// SelfAttentionAdaLNBlock_24292335026676
// MI455X (gfx1250) — compile-verified
//
#include <hip/hip_runtime.h>
#include <math.h>

// ---------------------------------------------------------------------------
// Problem constants (match reference)
// ---------------------------------------------------------------------------
#define BATCH 32768
#define DIM   1024
#define NHEAD 16
#define HDIM  64
#define LN_EPS 1e-5f

typedef __attribute__((ext_vector_type(2))) float v2f;
typedef __attribute__((ext_vector_type(8))) float v8f;
typedef __attribute__((ext_vector_type(4))) int   v4i;

// ---------------------------------------------------------------------------
// Device-side feature detection (device pass only; host pass gets fallback
// body so host & device agree on kernel symbols).
// ---------------------------------------------------------------------------
#if defined(__HIP_DEVICE_COMPILE__) && \
    __has_builtin(__builtin_amdgcn_global_load_async_to_lds_b128) && \
    __has_builtin(__builtin_amdgcn_s_wait_asynccnt)
#define USE_ASYNC_LDS 1
#else
#define USE_ASYNC_LDS 0
#endif

#if USE_ASYNC_LDS
typedef __attribute__((address_space(1))) v4i gl_v4i_t;
typedef __attribute__((address_space(3))) v4i lds_v4i_t;
__device__ __forceinline__ void async_cp16(const float* g, float* l) {
    __builtin_amdgcn_global_load_async_to_lds_b128(
        (gl_v4i_t*)g, (lds_v4i_t*)l, 0, 0);
}
#endif

__device__ __forceinline__ float fast_rcp(float x) {
#if defined(__HIP_DEVICE_COMPILE__) && __has_builtin(__builtin_amdgcn_rcpf)
    return __builtin_amdgcn_rcpf(x);
#else
    return 1.0f / x;
#endif
}
__device__ __forceinline__ float silu_f(float v) {
    return v * fast_rcp(1.0f + __expf(-v));
}
__device__ __forceinline__ float gelu_exact_f(float v) {
    return 0.5f * v * (1.0f + erff(v * 0.70710678118654752f));
}

// ---------------------------------------------------------------------------
// Tiled fp32 GEMM using V_WMMA_F32_16X16X4_F32.
//   C[M,N] = epilogue(A[M,K] @ Bw[K,N] + bias[N])
// 256 threads = 8 waves. Block tile 128x128, K-chunk 16.
// Wave grid 2(M) x 4(N): each wave computes 64x32 = 4x2 wmma tiles.
// Async path: double-buffered GLOBAL_LOAD_ASYNC_TO_LDS_B128, 1 barrier/chunk.
// EPI: 0 = none, 1 = +res residual, 2 = exact GELU.
// ---------------------------------------------------------------------------
template <int EPI>
__global__ __launch_bounds__(256)
void gemm_wmma(const float* __restrict__ A,
               const float* __restrict__ Bw,
               const float* __restrict__ bias,
               const float* __restrict__ res,
               float* __restrict__ C,
               int N, int K, int lda, int ldb) {
    const int tid  = threadIdx.x;
    const int lane = tid & 31;
    const int wave = tid >> 5;
    const int g    = lane >> 4;     // half-wave K-offset selector
    const int m16  = lane & 15;
    const int wm   = wave >> 2;     // 0..1 -> M offset wm*64
    const int wn   = wave & 3;      // 0..3 -> N offset wn*32

    const int row0 = blockIdx.y * 128;
    const int col0 = blockIdx.x * 128;

    const int ar = tid >> 1;        // 0..127  A-tile row
    const int ac = (tid & 1) * 8;   // 0|8     A-tile k-offset
    const int br = tid >> 4;        // 0..15   B-tile k-row
    const int bc = (tid & 15) * 8;  // 0..120  B-tile col

    v8f acc[4][2];
#pragma unroll
    for (int i = 0; i < 4; ++i)
#pragma unroll
        for (int j = 0; j < 2; ++j)
            acc[i][j] = (v8f){0.f,0.f,0.f,0.f,0.f,0.f,0.f,0.f};

#if USE_ASYNC_LDS
    // ---------------- async double-buffered pipeline ----------------
    __shared__ __align__(16) float As[2][128][20];  // [m][k], 80B rows
    __shared__ __align__(16) float Bs[2][16][136];  // [k][n], 544B rows

    auto stage = [&](int kk, int b) {
        const float* ap = A + (size_t)(row0 + ar) * lda + kk + ac;
        async_cp16(ap,     &As[b][ar][ac]);
        async_cp16(ap + 4, &As[b][ar][ac + 4]);
        const float* bp = Bw + (size_t)(kk + br) * ldb + col0 + bc;
        async_cp16(bp,     &Bs[b][br][bc]);
        async_cp16(bp + 4, &Bs[b][br][bc + 4]);
    };

    stage(0, 0);
    int buf = 0;
    for (int kk = 0; kk < K; kk += 16) {
        __builtin_amdgcn_s_wait_asynccnt(0);  // this wave's copies for buf done
        __syncthreads();                      // all waves' copies done; prev compute done
        if (kk + 16 < K) stage(kk + 16, buf ^ 1);

#pragma unroll
        for (int t = 0; t < 4; ++t) {
            const int kf = 4 * t + 2 * g;
            v2f af[4], bf[2];
#pragma unroll
            for (int i = 0; i < 4; ++i)
                af[i] = *(const v2f*)&As[buf][wm * 64 + i * 16 + m16][kf];
#pragma unroll
            for (int j = 0; j < 2; ++j) {
                const int n = wn * 32 + j * 16 + m16;
                bf[j].x = Bs[buf][kf][n];
                bf[j].y = Bs[buf][kf + 1][n];
            }
#pragma unroll
            for (int i = 0; i < 4; ++i)
#pragma unroll
                for (int j = 0; j < 2; ++j)
                    acc[i][j] = __builtin_amdgcn_wmma_f32_16x16x4_f32(
                        false, af[i], false, bf[j], (short)0, acc[i][j],
                        false, false);
        }
        buf ^= 1;
    }
#else
    // ---------------- synchronous fallback (proven path) ----------------
    __shared__ float As[128][18];
    __shared__ float Bs[128][18];   // [n][k] transposed

    for (int kk = 0; kk < K; kk += 16) {
        const float* ap = A + (size_t)(row0 + ar) * lda + kk + ac;
        float4 av0 = *(const float4*)(ap);
        float4 av1 = *(const float4*)(ap + 4);
        const float* bp = Bw + (size_t)(kk + br) * ldb + col0 + bc;
        float4 bv0 = *(const float4*)(bp);
        float4 bv1 = *(const float4*)(bp + 4);

        __syncthreads();
        As[ar][ac + 0] = av0.x; As[ar][ac + 1] = av0.y;
        As[ar][ac + 2] = av0.z; As[ar][ac + 3] = av0.w;
        As[ar][ac + 4] = av1.x; As[ar][ac + 5] = av1.y;
        As[ar][ac + 6] = av1.z; As[ar][ac + 7] = av1.w;
        Bs[bc + 0][br] = bv0.x; Bs[bc + 1][br] = bv0.y;
        Bs[bc + 2][br] = bv0.z; Bs[bc + 3][br] = bv0.w;
        Bs[bc + 4][br] = bv1.x; Bs[bc + 5][br] = bv1.y;
        Bs[bc + 6][br] = bv1.z; Bs[bc + 7][br] = bv1.w;
        __syncthreads();

#pragma unroll
        for (int t = 0; t < 4; ++t) {
            const int kf = 4 * t + 2 * g;
            v2f af[4], bf[2];
#pragma unroll
            for (int i = 0; i < 4; ++i)
                af[i] = *(const v2f*)&As[wm * 64 + i * 16 + m16][kf];
#pragma unroll
            for (int j = 0; j < 2; ++j)
                bf[j] = *(const v2f*)&Bs[wn * 32 + j * 16 + m16][kf];
#pragma unroll
            for (int i = 0; i < 4; ++i)
#pragma unroll
                for (int j = 0; j < 2; ++j)
                    acc[i][j] = __builtin_amdgcn_wmma_f32_16x16x4_f32(
                        false, af[i], false, bf[j], (short)0, acc[i][j],
                        false, false);
        }
    }
#endif

    // ---- epilogue ----
#pragma unroll
    for (int i = 0; i < 4; ++i) {
        const int basem = row0 + wm * 64 + i * 16;
#pragma unroll
        for (int j = 0; j < 2; ++j) {
            const int cn = col0 + wn * 32 + j * 16 + m16;
            const float bv = bias[cn];
#pragma unroll
            for (int v = 0; v < 8; ++v) {
                const int r = basem + v + 8 * g;
                float val = acc[i][j][v] + bv;
                if (EPI == 2) val = gelu_exact_f(val);
                if (EPI == 1) val += res[(size_t)r * N + cn];
                C[(size_t)r * N + cn] = val;
            }
        }
    }
}

// ---------------------------------------------------------------------------
// Elementwise SiLU (one pass over cond), float4 vectorized.
// ---------------------------------------------------------------------------
__global__ __launch_bounds__(256)
void silu_kernel(const float* __restrict__ in, float* __restrict__ out) {
    const size_t i = (size_t)blockIdx.x * 256 + threadIdx.x;
    float4 v = ((const float4*)in)[i];
    float4 r;
    r.x = silu_f(v.x); r.y = silu_f(v.y);
    r.z = silu_f(v.z); r.w = silu_f(v.w);
    ((float4*)out)[i] = r;
}

// ---------------------------------------------------------------------------
// LayerNorm (no affine) + AdaLN modulation:
//   out = (1 + ada[row, goff+c]) * LN(x)[row,c] + ada[row, moff+c]
// ---------------------------------------------------------------------------
__global__ __launch_bounds__(256)
void ln_modulate(const float* __restrict__ x,
                 const float* __restrict__ ada,
                 int goff, int moff,
                 float* __restrict__ out) {
    const int row = blockIdx.x;
    const int t = threadIdx.x;
    const float* xr = x + (size_t)row * DIM;

    float v[4];
    float s = 0.f, ss = 0.f;
#pragma unroll
    for (int i = 0; i < 4; ++i) {
        v[i] = xr[t + i * 256];
        s += v[i];
        ss += v[i] * v[i];
    }
    __shared__ float sm[256];
    __shared__ float sq[256];
    sm[t] = s; sq[t] = ss;
    __syncthreads();
    for (int off = 128; off > 0; off >>= 1) {
        if (t < off) { sm[t] += sm[t + off]; sq[t] += sq[t + off]; }
        __syncthreads();
    }
    const float mean = sm[0] * (1.0f / DIM);
    const float var  = sq[0] * (1.0f / DIM) - mean * mean;
    const float rs   = rsqrtf(var + LN_EPS);

    const float* ga = ada + (size_t)row * (4 * DIM) + goff;
    const float* mu = ada + (size_t)row * (4 * DIM) + moff;
    float* orow = out + (size_t)row * DIM;
#pragma unroll
    for (int i = 0; i < 4; ++i) {
        const int c = t + i * 256;
        orow[c] = (1.0f + ga[c]) * ((v[i] - mean) * rs) + mu[c];
    }
}

// ---------------------------------------------------------------------------
// Two-token attention, output at position 0 only. One wave32 per (row, head).
// Qx = [B, 3D] (q|k|v of x-token), KVc = [B, 2D] (k|v of cond token).
// ---------------------------------------------------------------------------
__global__ __launch_bounds__(256)
void attn_2tok(const float* __restrict__ Qx,
               const float* __restrict__ KVc,
               float* __restrict__ ctx) {
    const int wave = threadIdx.x >> 5;
    const int lane = threadIdx.x & 31;
    const int idx  = blockIdx.x * 8 + wave;   // row*NHEAD + head
    const int b = idx >> 4;
    const int h = idx & 15;

    const float* qp = Qx + (size_t)b * (3 * DIM) + h * HDIM;
    const float* k0 = qp + DIM;
    const float* v0 = qp + 2 * DIM;
    const float* k1 = KVc + (size_t)b * (2 * DIM) + h * HDIM;
    const float* v1 = k1 + DIM;

    const float q0 = qp[lane], q1 = qp[lane + 32];
    float s0 = q0 * k0[lane] + q1 * k0[lane + 32];
    float s1 = q0 * k1[lane] + q1 * k1[lane + 32];
#pragma unroll
    for (int m = 16; m >= 1; m >>= 1) {
        s0 += __shfl_xor(s0, m, 32);
        s1 += __shfl_xor(s1, m, 32);
    }
    s0 *= 0.125f;   // 1/sqrt(64)
    s1 *= 0.125f;
    const float mx = fmaxf(s0, s1);
    const float e0 = __expf(s0 - mx);
    const float e1 = __expf(s1 - mx);
    const float inv = 1.0f / (e0 + e1);
    const float a0 = e0 * inv, a1 = e1 * inv;

    float* cp = ctx + (size_t)b * DIM + h * HDIM;
    cp[lane]      = a0 * v0[lane]      + a1 * v1[lane];
    cp[lane + 32] = a0 * v0[lane + 32] + a1 * v1[lane + 32];
}

// ---------------------------------------------------------------------------
// Launch
// ---------------------------------------------------------------------------
extern "C" void kernel_launch(void* const* d_in, const int* in_sizes, int n_in,
                              void* d_out, int out_size, void* d_ws, size_t ws_size,
                              hipStream_t stream) {
    const float* x    = (const float*)d_in[0];
    const float* cond = (const float*)d_in[1];
    const float* Wqkv = (const float*)d_in[2];
    const float* bqkv = (const float*)d_in[3];
    const float* Wout = (const float*)d_in[4];
    const float* bout = (const float*)d_in[5];
    const float* W1   = (const float*)d_in[6];
    const float* b1   = (const float*)d_in[7];
    const float* W2   = (const float*)d_in[8];
    const float* b2   = (const float*)d_in[9];
    const float* Wada = (const float*)d_in[10];
    const float* bada = (const float*)d_in[11];
    float* out = (float*)d_out;

    float* ws  = (float*)d_ws;
    float* ada = ws;                                   // [B, 4D]
    float* Qb  = ada + (size_t)BATCH * 4 * DIM;        // [B, 3D] qkv of x-token
    float* K2  = Qb  + (size_t)BATCH * 3 * DIM;        // [B, 2D] kv of cond token
    float* T0  = K2  + (size_t)BATCH * 2 * DIM;        // [B, D] xn / ctx / xn2
    float* T1  = T0  + (size_t)BATCH * DIM;            // [B, D] x1
    float* T2  = T1  + (size_t)BATCH * DIM;            // [B, D] silu(cond)
    float* Hb  = Qb;                                   // [B, 4D] MLP hidden (reuse)

    const dim3 blk(256);
    const int MB = BATCH / 128;   // 256 row-blocks

    // 0) sc = silu(cond)
    silu_kernel<<<(BATCH * DIM) / (256 * 4), blk, 0, stream>>>(cond, T2);

    // 1) ada = sc @ Wada + bada
    gemm_wmma<0><<<dim3(4 * DIM / 128, MB), blk, 0, stream>>>(
        T2, Wada, bada, nullptr, ada, 4 * DIM, DIM, DIM, 4 * DIM);

    // 2) xn = (1+gamma1)*LN(x) + mu1
    ln_modulate<<<BATCH, blk, 0, stream>>>(x, ada, 0, DIM, T0);

    // 3) qkv of x-token
    gemm_wmma<0><<<dim3(3 * DIM / 128, MB), blk, 0, stream>>>(
        T0, Wqkv, bqkv, nullptr, Qb, 3 * DIM, DIM, DIM, 3 * DIM);

    // 4) k,v of cond token (columns D..3D of Wqkv)
    gemm_wmma<0><<<dim3(2 * DIM / 128, MB), blk, 0, stream>>>(
        cond, Wqkv + DIM, bqkv + DIM, nullptr, K2, 2 * DIM, DIM, DIM, 3 * DIM);

    // 5) attention over the two tokens, output at position 0
    attn_2tok<<<BATCH * NHEAD / 8, blk, 0, stream>>>(Qb, K2, T0);

    // 6) x1 = x + ctx @ Wout + bout
    gemm_wmma<1><<<dim3(DIM / 128, MB), blk, 0, stream>>>(
        T0, Wout, bout, x, T1, DIM, DIM, DIM, DIM);

    // 7) xn2 = (1+gamma2)*LN(x1) + mu2
    ln_modulate<<<BATCH, blk, 0, stream>>>(T1, ada, 2 * DIM, 3 * DIM, T0);

    // 8) h = gelu(xn2 @ W1 + b1)
    gemm_wmma<2><<<dim3(4 * DIM / 128, MB), blk, 0, stream>>>(
        T0, W1, b1, nullptr, Hb, 4 * DIM, DIM, DIM, 4 * DIM);

    // 9) out = x1 + h @ W2 + b2
    gemm_wmma<1><<<dim3(DIM / 128, MB), blk, 0, stream>>>(
        Hb, W2, b2, T1, out, DIM, 4 * DIM, 4 * DIM, DIM);
}